// AdaptiveFeaturePooling_60507499266508
// MI455X (gfx1250) — compile-verified
//
#include <hip/hip_runtime.h>
#include <stdint.h>

#define NOUT   14
#define GG     28              // sample grid per axis (NOUT * sr)
#define NS     (GG * GG)       // 784 samples
#define NPIX   (NOUT * NOUT)   // 196 output pixels
#define NCH    256
#define THREADS 256

typedef unsigned int u32;
typedef __attribute__((ext_vector_type(4))) u32 v4u32;
typedef __attribute__((ext_vector_type(8))) int v8i32;
typedef __attribute__((ext_vector_type(4))) int v4i32;

__global__ __launch_bounds__(THREADS)
void roi_fpn_maxpool_kernel(const float* __restrict__ f0,
                            const float* __restrict__ f1,
                            const float* __restrict__ f2,
                            const float* __restrict__ f3,
                            const float* __restrict__ rois,
                            float* __restrict__ out)
{
    __shared__ float4 s_w[NS];                  // 4 bilinear weights per sample (mask folded in)
    __shared__ int4   s_id[NS];                 // 4 tap indices per sample
    __shared__ __align__(16) float s_box[4];    // this ROI's (x1,y1,x2,y2)

    const int roi  = blockIdx.x;
    const int tid  = threadIdx.x;
    const int wave = tid >> 5;
    const int lane = tid & 31;

    // ---- Stage the ROI box into LDS via the Tensor Data Mover (async DMA) ----
#if defined(__HIP_DEVICE_COMPILE__) && __has_builtin(__builtin_amdgcn_tensor_load_to_lds) && __has_builtin(__builtin_amdgcn_s_wait_tensorcnt)
    if (tid == 0) {
        const u32 lds_off = (u32)(uintptr_t)&s_box[0];
        const uint64_t ga = (uint64_t)(uintptr_t)(rois + (size_t)roi * 4);
        v4u32 g0; v8i32 g1; v4i32 gz4; v8i32 gz8;
        g0[0] = 1u;                                            // count=1, user mode
        g0[1] = lds_off;                                       // lds_addr
        g0[2] = (u32)ga;                                       // global_addr[31:0]
        g0[3] = (u32)((ga >> 32) & 0x01FFFFFFu) | (2u << 30);  // global_addr[56:32] | type=2
        g1[0] = (2 << 16);                                     // data_size = 4 bytes
        g1[1] = (4 << 16);                                     // tensor_dim0 = 4 (bits 79:48)
        g1[2] = (1 << 16);                                     // tensor_dim1 = 1 (bits 111:80)
        g1[3] = (4 << 16);                                     // tile_dim0 = 4 (bits 127:112)
        g1[4] = 1;                                             // tile_dim1 = 1
        g1[5] = 4;                                             // tensor_dim0_stride = 4
        g1[6] = (4 << 16);                                     // tensor_dim1_stride = 4
        g1[7] = 0;
        gz4[0] = gz4[1] = gz4[2] = gz4[3] = 0;
        gz8[0] = gz8[1] = gz8[2] = gz8[3] = 0;
        gz8[4] = gz8[5] = gz8[6] = gz8[7] = 0;
        __builtin_amdgcn_tensor_load_to_lds(g0, g1, gz4, gz4, gz8, 0);
        __builtin_amdgcn_s_wait_tensorcnt(0);
    }
#else
    if (tid < 4) s_box[tid] = rois[(size_t)roi * 4 + tid];
#endif
    __syncthreads();

    const float rx1 = s_box[0], ry1 = s_box[1], rx2 = s_box[2], ry2 = s_box[3];

    // Channel blocks of 32: 8 waves x 4 channels each. Running max lives in VGPRs
    // so the 103MB output is written to HBM exactly once (no RMW across levels).
#pragma unroll 1
    for (int cb = 0; cb < NCH / 32; ++cb) {
        const int cbase = cb * 32 + wave * 4;

        float acc[4][7];
#pragma unroll
        for (int i = 0; i < 4; ++i)
#pragma unroll
            for (int j = 0; j < 7; ++j) acc[i][j] = -__builtin_inff();

        // Reference applies the scale cumulatively: i=3..0, r *= 2^i * 28.
        float x1 = rx1, y1 = ry1, x2 = rx2, y2 = ry2;

#pragma unroll 1
        for (int lv = 0; lv < 4; ++lv) {        // lv=0 -> i=3 (feat3, 28x28) ... lv=3 -> i=0 (feat0, 224x224)
            const float m = (float)(224 >> lv); // 224, 112, 56, 28
            x1 *= m; y1 *= m; x2 *= m; y2 *= m;

            const int   L  = 28 << lv;          // 28, 56, 112, 224
            const float Lf = (float)L;
            const int   HW = L * L;
            const float* __restrict__ f = (lv == 0) ? f3 : (lv == 1) ? f2 : (lv == 2) ? f1 : f0;

            __syncthreads();                    // LDS tables about to be overwritten

            // ---- Phase A: per-sample bilinear tables (channel-independent) ----
            const float rw = fmaxf(x2 - x1, 1.0f);
            const float rh = fmaxf(y2 - y1, 1.0f);
            const float sx = rw / (float)NOUT;
            const float sy = rh / (float)NOUT;
            for (int s = tid; s < NS; s += THREADS) {
                const int gy = s / GG, gx = s - gy * GG;
                float cy = y1 + ((float)gy + 0.5f) * 0.5f * sy;
                float cx = x1 + ((float)gx + 0.5f) * 0.5f * sx;
                const bool vy = (cy >= -1.0f) && (cy <= Lf);
                const bool vx = (cx >= -1.0f) && (cx <= Lf);
                cy = fminf(fmaxf(cy, 0.0f), Lf - 1.0f);
                cx = fminf(fmaxf(cx, 0.0f), Lf - 1.0f);
                const float fly = floorf(cy), flx = floorf(cx);
                float fy = cy - fly, fx = cx - flx;
                int yl = (int)fly, xl = (int)flx;
                int yh = yl + 1,   xh = xl + 1;
                if (yl >= L - 1) { yl = L - 1; yh = L - 1; fy = 0.0f; }
                if (xl >= L - 1) { xl = L - 1; xh = L - 1; fx = 0.0f; }
                const float msk = (vy && vx) ? 1.0f : 0.0f;
                const float oy1 = 1.0f - fy, ox1 = 1.0f - fx;
                s_w[s]  = make_float4(oy1 * ox1 * msk, oy1 * fx * msk,
                                      fy  * ox1 * msk, fy  * fx * msk);
                s_id[s] = make_int4(yl * L + xl, yl * L + xh,
                                    yh * L + xl, yh * L + xh);
            }
            __syncthreads();

            // ---- Phase B: gather + weighted accumulate, lanes = pixels ----
            const float* __restrict__ base = f + (size_t)cbase * HW;
            __builtin_prefetch(base, 0, 1);     // global_prefetch_b8: pull channel slab toward L2
#pragma unroll
            for (int pc = 0; pc < 7; ++pc) {
                const int p = pc * 32 + lane;
                if (p < NPIX) {
                    const int oy = p / NOUT, ox = p - oy * NOUT;
                    float sum0 = 0.f, sum1 = 0.f, sum2 = 0.f, sum3 = 0.f;
#pragma unroll
                    for (int a = 0; a < 2; ++a) {
#pragma unroll
                        for (int b = 0; b < 2; ++b) {
                            const int s = (oy * 2 + a) * GG + (ox * 2 + b);
                            const float4 w = s_w[s];
                            // w.x == 0  <=>  sample fully masked (fx,fy < 1 always):
                            // skip all 16 gathers for out-of-range samples.
                            if (w.x > 0.0f) {
                                const int4 id = s_id[s];
                                const float* b0 = base;
                                const float* b1 = base + HW;
                                const float* b2 = base + 2 * (size_t)HW;
                                const float* b3 = base + 3 * (size_t)HW;
                                sum0 += w.x * b0[id.x] + w.y * b0[id.y] + w.z * b0[id.z] + w.w * b0[id.w];
                                sum1 += w.x * b1[id.x] + w.y * b1[id.y] + w.z * b1[id.z] + w.w * b1[id.w];
                                sum2 += w.x * b2[id.x] + w.y * b2[id.y] + w.z * b2[id.z] + w.w * b2[id.w];
                                sum3 += w.x * b3[id.x] + w.y * b3[id.y] + w.z * b3[id.z] + w.w * b3[id.w];
                            }
                        }
                    }
                    acc[0][pc] = fmaxf(acc[0][pc], 0.25f * sum0);
                    acc[1][pc] = fmaxf(acc[1][pc], 0.25f * sum1);
                    acc[2][pc] = fmaxf(acc[2][pc], 0.25f * sum2);
                    acc[3][pc] = fmaxf(acc[3][pc], 0.25f * sum3);
                }
            }
        }

        // ---- Store: coalesced, one HBM write per output element ----
#pragma unroll
        for (int c4 = 0; c4 < 4; ++c4) {
            float* __restrict__ po = out + ((size_t)roi * NCH + (cbase + c4)) * NPIX;
#pragma unroll
            for (int pc = 0; pc < 7; ++pc) {
                const int p = pc * 32 + lane;
                if (p < NPIX) po[p] = acc[c4][pc];
            }
        }
    }
}

extern "C" void kernel_launch(void* const* d_in, const int* in_sizes, int n_in,
                              void* d_out, int out_size, void* d_ws, size_t ws_size,
                              hipStream_t stream) {
    const float* f0   = (const float*)d_in[0];   // [1,256,224,224]
    const float* f1   = (const float*)d_in[1];   // [1,256,112,112]
    const float* f2   = (const float*)d_in[2];   // [1,256,56,56]
    const float* f3   = (const float*)d_in[3];   // [1,256,28,28]
    const float* rois = (const float*)d_in[4];   // [R,4]
    float* out = (float*)d_out;                  // [R,256,14,14]
    const int R = in_sizes[4] / 4;
    roi_fpn_maxpool_kernel<<<dim3(R), dim3(THREADS), 0, stream>>>(f0, f1, f2, f3, rois, out);
}